// TransportKernel_15625091023367
// MI455X (gfx1250) — compile-verified
//
#include <hip/hip_runtime.h>

typedef __attribute__((ext_vector_type(2))) float v2f;
typedef __attribute__((ext_vector_type(8))) float v8f;
typedef __attribute__((ext_vector_type(4))) int   v4i;

#define NN 1024
#define DD 128
#define NUGGET_F 1e-3f
#define REG_LAMBDA_F 1e-2f
#define CG_ITERS 64

#if defined(__has_builtin)
#if __has_builtin(__builtin_amdgcn_global_load_async_to_lds_b128)
#define ASYNC_LDS 1
#endif
#endif

// D = A(16x4,f32) x B(4x16,f32) + C(16x16,f32)   [V_WMMA_F32_16X16X4_F32]
__device__ __forceinline__ v8f wmma_f32_k4(v2f a, v2f b, v8f c) {
  return __builtin_amdgcn_wmma_f32_16x16x4_f32(
      /*neg_a=*/false, a, /*neg_b=*/false, b,
      /*c_mod=*/(short)0, c, /*reuse_a=*/false, /*reuse_b=*/false);
}

// 16B global -> LDS copy; async (ASYNCcnt-tracked) path when available.
// Builtin expects non-const v4i* operands (per compiler diagnostic).
__device__ __forceinline__ void copy_b128_to_lds(const float* g, float* l) {
#ifdef ASYNC_LDS
  __builtin_amdgcn_global_load_async_to_lds_b128(
      (__attribute__((address_space(1))) v4i*)(void*)const_cast<float*>(g),
      (__attribute__((address_space(3))) v4i*)(void*)l,
      /*offset=*/0, /*cpol=*/0);
#else
  *(float4*)l = *(const float4*)g;
#endif
}

__device__ __forceinline__ void lds_copy_join() {
#ifdef ASYNC_LDS
#if __has_builtin(__builtin_amdgcn_s_wait_asynccnt)
  __builtin_amdgcn_s_wait_asynccnt(0);
#endif
#endif
  __syncthreads();
}

// ---------------------------------------------------------------------------
// M = Z + X ; row norms ||x||^2, ||y||^2, ||m||^2.  One block (128 thr) / row.
// ---------------------------------------------------------------------------
__global__ void prep_kernel(const float* __restrict__ X, const float* __restrict__ Y,
                            const float* __restrict__ Z, float* __restrict__ M,
                            float* __restrict__ nx, float* __restrict__ ny,
                            float* __restrict__ nm) {
  const int row = blockIdx.x;
  const int t = threadIdx.x;  // 128
  const float x = X[row * DD + t];
  const float y = Y[row * DD + t];
  const float z = Z[row * DD + t];
  const float m = x + z;
  M[row * DD + t] = m;
  __shared__ float sx[DD], sy[DD], sm[DD];
  sx[t] = x * x; sy[t] = y * y; sm[t] = m * m;
  __syncthreads();
  for (int s = 64; s > 0; s >>= 1) {
    if (t < s) { sx[t] += sx[t + s]; sy[t] += sy[t + s]; sm[t] += sm[t + s]; }
    __syncthreads();
  }
  if (t == 0) { nx[row] = sx[0]; ny[row] = sy[0]; nm[row] = sm[0]; }
}

__global__ void zero_kernel(float* __restrict__ p, int n) {
  const int i = blockIdx.x * blockDim.x + threadIdx.x;
  if (i < n) p[i] = 0.0f;
}

// ---------------------------------------------------------------------------
// RBF Gram via fp32 WMMA: S = A.B^T then k = exp(-(na+nb-2S)/2).
// The 16 A-rows are shared by all 8 waves of a block -> staged once in LDS
// (async global->LDS when available).
// outMat != null : store full matrix (+ nugget on diagonal).
// outMat == null : fused reduction of the full-matrix sum into *outSum.
// ---------------------------------------------------------------------------
__global__ void gram_exp_kernel(const float* __restrict__ A, const float* __restrict__ B,
                                const float* __restrict__ na, const float* __restrict__ nb,
                                float* __restrict__ outMat, float* __restrict__ outSum,
                                int addNugget) {
  const int lane  = threadIdx.x & 31;
  const int wave  = threadIdx.x >> 5;
  const int tileM = blockIdx.x >> 3;                  // uniform per block
  const int tileN = (blockIdx.x & 7) * 8 + wave;
  const int row16 = lane & 15;
  const int ksel  = (lane >> 4) << 1;                 // lanes 16-31 hold K+2,K+3

  __shared__ __align__(16) float ldsA[16 * DD];       // 8 KB: 16 rows x 128
  {
    const int t   = threadIdx.x;                      // 256 threads, 8 floats each
    const int r   = t >> 4;                           // 0..15
    const int col = (t & 15) * 8;                     // 0,8,...,120
    const float* g = A + (size_t)(tileM * 16 + r) * DD + col;
    copy_b128_to_lds(g,     &ldsA[r * DD + col]);
    copy_b128_to_lds(g + 4, &ldsA[r * DD + col + 4]);
  }
  lds_copy_join();

  const float* __restrict__ brow = B + (size_t)(tileN * 16 + row16) * DD;
  const float* __restrict__ arow = &ldsA[row16 * DD];

  v8f acc = {};
  #pragma unroll 4
  for (int k0 = 0; k0 < DD; k0 += 4) {
    v2f a = *(const v2f*)(arow + k0 + ksel);          // ds_load_b64
    v2f b = *(const v2f*)(brow + k0 + ksel);          // global b64
    acc = wmma_f32_k4(a, b, acc);
  }

  const int mbase = tileM * 16 + ((lane >> 4) << 3);  // C: lane[4] -> M or M+8
  const int gn    = tileN * 16 + row16;
  const float nbv = nb[gn];
  if (outMat) {
    #pragma unroll
    for (int r = 0; r < 8; ++r) {
      const int gm = mbase + r;
      float v = __expf(-0.5f * (na[gm] + nbv - 2.0f * acc[r]));
      if (addNugget && gm == gn) v += NUGGET_F;
      outMat[(size_t)gm * NN + gn] = v;
    }
  } else {
    float s = 0.0f;
    #pragma unroll
    for (int r = 0; r < 8; ++r) {
      const int gm = mbase + r;
      s += __expf(-0.5f * (na[gm] + nbv - 2.0f * acc[r]));
    }
    for (int off = 16; off > 0; off >>= 1) s += __shfl_down(s, off, 32);
    if (lane == 0) atomicAdd(outSum, s);
  }
}

// ---------------------------------------------------------------------------
// Q = Kxx(1024x1024) @ P(1024x128) with fp32 WMMA.
// Block = 8 waves sharing one 16-row stripe of Kxx; A panels staged through
// LDS in K-chunks of 64 (async global->LDS), B streamed from global.
// ---------------------------------------------------------------------------
#define KC 64
__global__ void matvec_kernel(const float* __restrict__ K, const float* __restrict__ P,
                              float* __restrict__ Q) {
  const int lane  = threadIdx.x & 31;
  const int wave  = threadIdx.x >> 5;
  const int tileM = blockIdx.x;                       // 64 blocks
  const int tileN = wave;                             // 8 column tiles
  const int row16 = lane & 15;
  const int ksel  = (lane >> 4) << 1;
  const int gn    = tileN * 16 + row16;

  __shared__ __align__(16) float ldsA[16 * KC];       // 4 KB: 16 rows x 64
  const int tr = threadIdx.x >> 4;                    // 0..15
  const int tk = (threadIdx.x & 15) * 4;              // 0,4,...,60
  const float* __restrict__ krow = K + (size_t)(tileM * 16 + tr) * NN + tk;

  v8f acc = {};
  for (int k0 = 0; k0 < NN; k0 += KC) {
    copy_b128_to_lds(krow + k0, &ldsA[tr * KC + tk]); // one b128 per thread
    lds_copy_join();
    const float* __restrict__ arow = &ldsA[row16 * KC];
    #pragma unroll 4
    for (int kk = 0; kk < KC; kk += 4) {
      v2f a = *(const v2f*)(arow + kk + ksel);        // ds_load_b64
      v2f b;
      b.x = P[(size_t)(k0 + kk + ksel)     * DD + gn];
      b.y = P[(size_t)(k0 + kk + ksel + 1) * DD + gn];
      acc = wmma_f32_k4(a, b, acc);
    }
    __syncthreads();                                  // LDS reuse fence
  }
  const int mbase = tileM * 16 + ((lane >> 4) << 3);
  #pragma unroll
  for (int r = 0; r < 8; ++r)
    Q[(size_t)(mbase + r) * DD + gn] = acc[r];
}

// ---------------------------------------------------------------------------
// CG over 128 independent RHS columns: one block per column.
// ---------------------------------------------------------------------------
__global__ void cg_init_kernel(const float* __restrict__ Z, float* __restrict__ W,
                               float* __restrict__ R, float* __restrict__ P,
                               float* __restrict__ rr) {
  const int j = blockIdx.x;     // column 0..127
  const int t = threadIdx.x;    // 256
  __shared__ float red[256];
  float s = 0.0f;
  for (int i = t; i < NN; i += 256) {
    const float z = Z[(size_t)i * DD + j];
    W[(size_t)i * DD + j] = 0.0f;
    R[(size_t)i * DD + j] = z;
    P[(size_t)i * DD + j] = z;
    s += z * z;
  }
  red[t] = s; __syncthreads();
  for (int k = 128; k > 0; k >>= 1) { if (t < k) red[t] += red[t + k]; __syncthreads(); }
  if (t == 0) rr[j] = red[0];
}

__global__ void cg_update_kernel(const float* __restrict__ Q, float* __restrict__ W,
                                 float* __restrict__ R, float* __restrict__ P,
                                 float* __restrict__ rr) {
  const int j = blockIdx.x;
  const int t = threadIdx.x;    // 256
  __shared__ float red[256];
  __shared__ float s_alpha, s_beta;
  const float rr_old = rr[j];

  float pq = 0.0f;
  for (int i = t; i < NN; i += 256)
    pq += P[(size_t)i * DD + j] * Q[(size_t)i * DD + j];
  red[t] = pq; __syncthreads();
  for (int k = 128; k > 0; k >>= 1) { if (t < k) red[t] += red[t + k]; __syncthreads(); }
  if (t == 0) { const float d = red[0]; s_alpha = (d != 0.0f) ? rr_old / d : 0.0f; }
  __syncthreads();
  const float alpha = s_alpha;

  float rrn = 0.0f;
  for (int i = t; i < NN; i += 256) {
    const size_t idx = (size_t)i * DD + j;
    const float p  = P[idx];
    const float rn = R[idx] - alpha * Q[idx];
    W[idx] += alpha * p;
    R[idx]  = rn;
    rrn += rn * rn;
  }
  __syncthreads();
  red[t] = rrn; __syncthreads();
  for (int k = 128; k > 0; k >>= 1) { if (t < k) red[t] += red[t + k]; __syncthreads(); }
  if (t == 0) {
    const float rn2 = red[0];
    s_beta = (rr_old != 0.0f) ? rn2 / rr_old : 0.0f;
    rr[j] = rn2;
  }
  __syncthreads();
  const float beta = s_beta;
  for (int i = t; i < NN; i += 256) {
    const size_t idx = (size_t)i * DD + j;
    P[idx] = R[idx] + beta * P[idx];
  }
}

// ---------------------------------------------------------------------------
// loss = (sZZ - 2 sZY + sYY)/N^2 + lambda * sum(Z .* W)
// ---------------------------------------------------------------------------
__global__ void finalize_kernel(const float* __restrict__ Z, const float* __restrict__ W,
                                const float* __restrict__ sums, float* __restrict__ out) {
  const int t = threadIdx.x;    // 1024
  __shared__ float red[1024];
  float s = 0.0f;
  for (int i = t; i < NN * DD; i += 1024) s += Z[i] * W[i];
  red[t] = s; __syncthreads();
  for (int k = 512; k > 0; k >>= 1) { if (t < k) red[t] += red[t + k]; __syncthreads(); }
  if (t == 0) {
    const float invN2 = 1.0f / ((float)NN * (float)NN);
    const float mmd = (sums[0] - 2.0f * sums[1] + sums[2]) * invN2;
    out[0] = mmd + REG_LAMBDA_F * red[0];
  }
}

extern "C" void kernel_launch(void* const* d_in, const int* in_sizes, int n_in,
                              void* d_out, int out_size, void* d_ws, size_t ws_size,
                              hipStream_t stream) {
  (void)in_sizes; (void)n_in; (void)out_size; (void)ws_size;
  const float* X = (const float*)d_in[0];
  const float* Y = (const float*)d_in[1];
  const float* Z = (const float*)d_in[2];
  float* out = (float*)d_out;

  float* ws   = (float*)d_ws;
  float* M    = ws;                        // 1024*128
  float* nx   = M + NN * DD;               // 1024
  float* ny   = nx + NN;                   // 1024
  float* nm   = ny + NN;                   // 1024
  float* Kxx  = nm + NN;                   // 1024*1024
  float* W    = Kxx + (size_t)NN * NN;     // 1024*128
  float* R    = W + NN * DD;               // 1024*128
  float* P    = R + NN * DD;               // 1024*128
  float* Q    = P + NN * DD;               // 1024*128
  float* sums = Q + NN * DD;               // [0]=ZZ [1]=ZY [2]=YY
  float* rr   = sums + 4;                  // 128

  prep_kernel<<<NN, DD, 0, stream>>>(X, Y, Z, M, nx, ny, nm);
  zero_kernel<<<1, 32, 0, stream>>>(sums, 4);

  // Kernel matrix (stored, + nugget) and fused MMD scalar reductions.
  gram_exp_kernel<<<512, 256, 0, stream>>>(X, X, nx, nx, Kxx, nullptr, 1);
  gram_exp_kernel<<<512, 256, 0, stream>>>(M, M, nm, nm, nullptr, sums + 0, 0);
  gram_exp_kernel<<<512, 256, 0, stream>>>(M, Y, nm, ny, nullptr, sums + 1, 0);
  gram_exp_kernel<<<512, 256, 0, stream>>>(Y, Y, ny, ny, nullptr, sums + 2, 0);

  // CG solve Kxx W = Z (128 RHS).
  cg_init_kernel<<<DD, 256, 0, stream>>>(Z, W, R, P, rr);
  for (int it = 0; it < CG_ITERS; ++it) {
    matvec_kernel<<<64, 256, 0, stream>>>(Kxx, P, Q);
    cg_update_kernel<<<DD, 256, 0, stream>>>(Q, W, R, P, rr);
  }

  finalize_kernel<<<1, 1024, 0, stream>>>(Z, W, sums, out);
}